// BilinearSampler_29300266893747
// MI455X (gfx1250) — compile-verified
//
#include <hip/hip_runtime.h>

// Problem constants from the reference (fixed shapes).
#define BATCH 16
#define HEIGHT 256
#define WIDTH  256
#define CHANS  32
#define NPIX   (BATCH * HEIGHT * WIDTH)   // 1,048,576

typedef __attribute__((ext_vector_type(2))) float v2f;

__device__ __forceinline__ int clampi(int v, int lo, int hi) {
    return v < lo ? lo : (v > hi ? hi : v);
}

// ---- gfx1250 async global->LDS gather path (probe via __has_builtin) ------
#if defined(__gfx1250__) && __has_builtin(__builtin_amdgcn_global_load_async_to_lds_b32)
#define USE_ASYNC_GATHER 1
#else
#define USE_ASYNC_GATHER 0
#endif

#if USE_ASYNC_GATHER
// One lane gathers one float into its own LDS slot (per-lane VGPR addresses
// on both sides, per ISA 15.18.3 GLOBAL_LOAD_ASYNC_TO_LDS_B32).
// Builtin signature (per hipcc diagnostic): (int AS1*, int AS3*, imm, imm).
#define ASYNC_GATHER_B32(gp, lp)                                            \
    __builtin_amdgcn_global_load_async_to_lds_b32(                          \
        (__attribute__((address_space(1))) int*)(void*)(gp),                \
        (__attribute__((address_space(3))) int*)(void*)(lp), 0, 0)

#if __has_builtin(__builtin_amdgcn_s_wait_asynccnt)
#define WAIT_ASYNC(n) do { __builtin_amdgcn_s_wait_asynccnt(n);             \
                           __asm__ __volatile__("" ::: "memory"); } while (0)
#else
#define WAIT_ASYNC(n) __asm__ __volatile__("s_wait_asynccnt %0" :: "n"(n) : "memory")
#endif

// Compute taps addresses + weights for pixel p and launch the 4 async
// gathers into this wave's LDS stage buffer (4 taps x 32 lanes).
__device__ __forceinline__ void stage_issue(
    v2f g, int p, int lane, const float* __restrict__ image, float* lds_stage,
    float& wa, float& wb, float& wc, float& wd)
{
    const float x = (g.x + 1.0f) * (0.5f * (WIDTH  - 1));
    const float y = (g.y + 1.0f) * (0.5f * (HEIGHT - 1));
    const int x0 = (int)floorf(x);
    const int y0 = (int)floorf(y);
    const int x0c = clampi(x0,     0, WIDTH  - 1);
    const int x1c = clampi(x0 + 1, 0, WIDTH  - 1);
    const int y0c = clampi(y0,     0, HEIGHT - 1);
    const int y1c = clampi(y0 + 1, 0, HEIGHT - 1);

    const float wxa = (float)x1c - x;
    const float wxb = x - (float)x0c;
    const float wya = (float)y1c - y;
    const float wyb = y - (float)y0c;
    wa = wxa * wya;   // (y0c, x0c)
    wb = wxa * wyb;   // (y1c, x0c)
    wc = wxb * wya;   // (y0c, x1c)
    wd = wxb * wyb;   // (y1c, x1c)

    const float* img_b = image + (size_t)(p >> 16) * (HEIGHT * WIDTH * CHANS);
    const float* r0 = img_b + (size_t)(y0c * WIDTH) * CHANS;
    const float* r1 = img_b + (size_t)(y1c * WIDTH) * CHANS;
    ASYNC_GATHER_B32(r0 + x0c * CHANS + lane, lds_stage + 0 * 32 + lane);
    ASYNC_GATHER_B32(r1 + x0c * CHANS + lane, lds_stage + 1 * 32 + lane);
    ASYNC_GATHER_B32(r0 + x1c * CHANS + lane, lds_stage + 2 * 32 + lane);
    ASYNC_GATHER_B32(r1 + x1c * CHANS + lane, lds_stage + 3 * 32 + lane);
}
#endif // USE_ASYNC_GATHER

__global__ __launch_bounds__(256) void BilinearSampler_kernel(
    const float* __restrict__ image,   // (B,H,W,C) f32
    const float* __restrict__ grid,    // (B,H,W,2) f32
    float*       __restrict__ out,     // (B,H,W,C) f32
    int nwaves_total)
{
    const int lane = threadIdx.x & 31;                        // lane == channel
    const int wave = (int)((blockIdx.x * blockDim.x + threadIdx.x) >> 5);

#if USE_ASYNC_GATHER
    // Per-wave private double buffer: 2 stages x 4 taps x 32 lanes floats.
    __shared__ float taps[256 / 32][2][4 * 32];               // 8 KB / block
    const int wib = threadIdx.x >> 5;
    float* buf0 = &taps[wib][0][0];
    float* buf1 = &taps[wib][1][0];

    int p = wave;
    if (p >= NPIX) return;

    // Prologue: issue stage 0's gathers.
    v2f g = __builtin_nontemporal_load((const v2f*)grid + p);
    float wa0, wb0, wc0, wd0;
    stage_issue(g, p, lane, image, buf0, wa0, wb0, wc0, wd0);

    int stage = 0;
    while (true) {
        const int  pn = p + nwaves_total;
        const bool hn = (pn < NPIX);

        // Issue next stage's 4 async gathers before consuming the current one.
        float wa1, wb1, wc1, wd1;
        if (hn) {
            const v2f gn = __builtin_nontemporal_load((const v2f*)grid + pn);
            stage_issue(gn, pn, lane, image, stage ? buf0 : buf1,
                        wa1, wb1, wc1, wd1);
        }

        // In-order async completion: <=4 outstanding means the older stage's
        // 4 gathers have landed in LDS.
        if (hn) WAIT_ASYNC(4); else WAIT_ASYNC(0);

        const float* cur = stage ? buf1 : buf0;
        const float Ia = cur[0 * 32 + lane];
        const float Ib = cur[1 * 32 + lane];
        const float Ic = cur[2 * 32 + lane];
        const float Id = cur[3 * 32 + lane];

        float acc = wa0 * Ia;
        acc = fmaf(wb0, Ib, acc);
        acc = fmaf(wc0, Ic, acc);
        acc = fmaf(wd0, Id, acc);
        // Output is write-once: NT store keeps the image resident in L2.
        __builtin_nontemporal_store(acc, out + (size_t)p * CHANS + lane);

        if (!hn) break;
        wa0 = wa1; wb0 = wb1; wc0 = wc1; wd0 = wd1;
        p = pn;
        stage ^= 1;
    }

#else  // -------- fallback: direct loads + global_prefetch_b8 pipeline ------
    int p = wave;
    if (p >= NPIX) return;

    v2f g = __builtin_nontemporal_load((const v2f*)grid + p);

    while (true) {
        const int  pn       = p + nwaves_total;
        const bool has_next = (pn < NPIX);

        v2f gn;
        if (has_next) gn = __builtin_nontemporal_load((const v2f*)grid + pn);

        const int b = p >> 16;

        float x = (g.x + 1.0f) * (0.5f * (WIDTH  - 1));
        float y = (g.y + 1.0f) * (0.5f * (HEIGHT - 1));

        const int x0 = (int)floorf(x);
        const int y0 = (int)floorf(y);
        const int x0c = clampi(x0,     0, WIDTH  - 1);
        const int x1c = clampi(x0 + 1, 0, WIDTH  - 1);
        const int y0c = clampi(y0,     0, HEIGHT - 1);
        const int y1c = clampi(y0 + 1, 0, HEIGHT - 1);

        const float wxa = (float)x1c - x;
        const float wxb = x - (float)x0c;
        const float wya = (float)y1c - y;
        const float wyb = y - (float)y0c;
        const float wa = wxa * wya;
        const float wb = wxa * wyb;
        const float wc = wxb * wya;
        const float wd = wxb * wyb;

        const float* img_b = image + (size_t)b * (HEIGHT * WIDTH * CHANS);
        const float* r0 = img_b + (size_t)(y0c * WIDTH) * CHANS;
        const float* r1 = img_b + (size_t)(y1c * WIDTH) * CHANS;
        const float Ia = r0[x0c * CHANS + lane];
        const float Ic = r0[x1c * CHANS + lane];
        const float Ib = r1[x0c * CHANS + lane];
        const float Id = r1[x1c * CHANS + lane];

        if (has_next) {
            const float xn = (gn.x + 1.0f) * (0.5f * (WIDTH  - 1));
            const float yn = (gn.y + 1.0f) * (0.5f * (HEIGHT - 1));
            const int nx0 = (int)floorf(xn);
            const int ny0 = (int)floorf(yn);
            const int nx0c = clampi(nx0,     0, WIDTH  - 1);
            const int nx1c = clampi(nx0 + 1, 0, WIDTH  - 1);
            const int ny0c = clampi(ny0,     0, HEIGHT - 1);
            const int ny1c = clampi(ny0 + 1, 0, HEIGHT - 1);
            const float* nimg_b = image + (size_t)(pn >> 16) * (HEIGHT * WIDTH * CHANS);
            const float* nr0 = nimg_b + (size_t)(ny0c * WIDTH) * CHANS;
            const float* nr1 = nimg_b + (size_t)(ny1c * WIDTH) * CHANS;
            __builtin_prefetch(nr0 + nx0c * CHANS + lane, 0, 1);
            __builtin_prefetch(nr0 + nx1c * CHANS + lane, 0, 1);
            __builtin_prefetch(nr1 + nx0c * CHANS + lane, 0, 1);
            __builtin_prefetch(nr1 + nx1c * CHANS + lane, 0, 1);
        }

        float acc = wa * Ia;
        acc = fmaf(wb, Ib, acc);
        acc = fmaf(wc, Ic, acc);
        acc = fmaf(wd, Id, acc);
        __builtin_nontemporal_store(acc, out + (size_t)p * CHANS + lane);

        if (!has_next) break;
        g = gn;
        p = pn;
    }
#endif
}

extern "C" void kernel_launch(void* const* d_in, const int* in_sizes, int n_in,
                              void* d_out, int out_size, void* d_ws, size_t ws_size,
                              hipStream_t stream) {
    const float* image = (const float*)d_in[0];   // (16,256,256,32) f32
    const float* grid  = (const float*)d_in[1];   // (16,256,256,2)  f32
    float*       out   = (float*)d_out;           // (16,256,256,32) f32

    const int threads = 256;                      // 8 waves / block
    const int blocks  = 4096;                     // 32768 waves, 32 pixels each
    const int nwaves  = (blocks * threads) >> 5;

    BilinearSampler_kernel<<<dim3(blocks), dim3(threads), 0, stream>>>(
        image, grid, out, nwaves);
}